// PriorScaleShift_47467978556092
// MI455X (gfx1250) — compile-verified
//
#include <hip/hip_runtime.h>
#include <hip/hip_bf16.h>

#define VOCAB     32000
#define BATCH     1024
#define SEQ       512
#define BITWORDS  (VOCAB / 32)   // 1000 dwords = 4000 bytes
#define BLOCK     256            // 8 wave32s

typedef __attribute__((ext_vector_type(4))) float v4f;
typedef __attribute__((ext_vector_type(4))) int   v4i;
typedef __attribute__((address_space(1))) v4i* gv4i_ptr;  // global int4*
typedef __attribute__((address_space(3))) v4i* lv4i_ptr;  // LDS int4*

#if __has_builtin(__builtin_amdgcn_global_load_async_to_lds_b128)
#define HAVE_ASYNC_LDS 1
#endif

// ---------------------------------------------------------------------------
// Kernel 1: on[v] = exp(log_scale[v]) + shift[v]  (32000 elements, trivial)
// ---------------------------------------------------------------------------
__global__ void PriorScaleShift_precompute_on(const float* __restrict__ log_scale,
                                              const float* __restrict__ shift,
                                              float* __restrict__ on) {
  int v = blockIdx.x * blockDim.x + threadIdx.x;
  if (v < VOCAB) on[v] = expf(log_scale[v]) + shift[v];
}

// ---------------------------------------------------------------------------
// Kernel 2: one workgroup per batch row.
//   Stage x/mask row into LDS with gfx1250 async-to-LDS b128 loads (overlapped
//   with bitmap zeroing), OR presence bits into a 32000-bit LDS bitmap via
//   ds_or_b32, then stream out[b,v] = bit ? on[v] : shift[v] with NT b128
//   stores (131 MB streaming output = the HBM roofline term).
// ---------------------------------------------------------------------------
__global__ __launch_bounds__(BLOCK) void PriorScaleShift_presence_kernel(
    const int* __restrict__ x, const float* __restrict__ mask,
    const float* __restrict__ log_scale, const float* __restrict__ shift,
    const float* __restrict__ on_pre,  // may be nullptr -> compute exp inline
    float* __restrict__ out) {
  __shared__ unsigned int bitmap[BITWORDS];
#if defined(HAVE_ASYNC_LDS)
  __shared__ int   x_s[SEQ];
  __shared__ float m_s[SEQ];
#endif

  const int b   = blockIdx.x;
  const int tid = threadIdx.x;

  const int*   __restrict__ xrow = x    + (size_t)b * SEQ;
  const float* __restrict__ mrow = mask + (size_t)b * SEQ;

#if defined(HAVE_ASYNC_LDS)
  // 256 lanes x 16B = 4096B: lanes 0..127 -> x row (2048B), 128..255 -> mask row.
  // tid<128 is wave-uniform (waves 0-3 vs 4-7), so no intra-wave divergence.
  if (tid < 128) {
    __builtin_amdgcn_global_load_async_to_lds_b128(
        (gv4i_ptr)(xrow + tid * 4),
        (lv4i_ptr)(&x_s[tid * 4]), 0, 0);
  } else {
    const int t = tid - 128;
    __builtin_amdgcn_global_load_async_to_lds_b128(
        (gv4i_ptr)(mrow + t * 4),
        (lv4i_ptr)(&m_s[t * 4]), 0, 0);
  }
#endif

  // Zero the presence bitmap while the async copies are in flight.
  for (int i = tid; i < BITWORDS; i += BLOCK) bitmap[i] = 0u;

#if defined(HAVE_ASYNC_LDS)
#if __has_builtin(__builtin_amdgcn_s_wait_asynccnt)
  __builtin_amdgcn_s_wait_asynccnt(0);
#else
  asm volatile("s_wait_asynccnt 0" ::: "memory");
#endif
#endif
  __syncthreads();

  // Bit-scatter: presence[v] |= (mask > 0) for every token in the row.
  for (int s = tid; s < SEQ; s += BLOCK) {
#if defined(HAVE_ASYNC_LDS)
    const int   tok = x_s[s];
    const float m   = m_s[s];
#else
    const int   tok = xrow[s];
    const float m   = mrow[s];
#endif
    if (m > 0.0f) atomicOr(&bitmap[tok >> 5], 1u << (tok & 31));
  }
  __syncthreads();

  // Streaming writer: 8000 float4 per row, non-temporal b128 stores.
  float* __restrict__ orow = out + (size_t)b * VOCAB;
  const v4f* __restrict__ sh4v = (const v4f*)shift;
  const v4f* __restrict__ on4v = (const v4f*)on_pre;
  const v4f* __restrict__ ls4v = (const v4f*)log_scale;

  for (int i = tid; i < VOCAB / 4; i += BLOCK) {
    const int v = i * 4;                       // v % 4 == 0 -> 4 bits same word
    const unsigned bits = bitmap[v >> 5] >> (v & 31);
    const v4f sh = sh4v[i];
    v4f on;
    if (on_pre != nullptr) {
      on = on4v[i];
    } else {
      const v4f ls = ls4v[i];
      on.x = expf(ls.x) + sh.x; on.y = expf(ls.y) + sh.y;
      on.z = expf(ls.z) + sh.z; on.w = expf(ls.w) + sh.w;
    }
    v4f r;
    r.x = (bits & 1u) ? on.x : sh.x;
    r.y = (bits & 2u) ? on.y : sh.y;
    r.z = (bits & 4u) ? on.z : sh.z;
    r.w = (bits & 8u) ? on.w : sh.w;
    __builtin_nontemporal_store(r, (v4f*)orow + i);
  }
}

// ---------------------------------------------------------------------------
extern "C" void kernel_launch(void* const* d_in, const int* in_sizes, int n_in,
                              void* d_out, int out_size, void* d_ws, size_t ws_size,
                              hipStream_t stream) {
  const int*   x         = (const int*)d_in[0];
  const float* mask      = (const float*)d_in[1];
  const float* log_scale = (const float*)d_in[2];
  const float* shift     = (const float*)d_in[3];
  float*       out       = (float*)d_out;

  float* on_pre = nullptr;
  if (ws_size >= (size_t)VOCAB * sizeof(float)) {
    on_pre = (float*)d_ws;
    PriorScaleShift_precompute_on<<<(VOCAB + 255) / 256, 256, 0, stream>>>(
        log_scale, shift, on_pre);
  }

  PriorScaleShift_presence_kernel<<<BATCH, BLOCK, 0, stream>>>(
      x, mask, log_scale, shift, on_pre, out);
}